// ZCANormSVDunstable_45792941310323
// MI455X (gfx1250) — compile-verified
//
#include <hip/hip_runtime.h>
#include <cstddef>
#include <cstdint>

// ZCA group whitening (SVD variant), MI455X / gfx1250.
// FP32 end-to-end; GEMM phases use V_WMMA_F32_16X16X4_F32 (wave32 WMMA).
// Single-pass mean+cov (cov = S/M - mu*mu^T): ~384 MiB HBM traffic
// -> ~16.5 us floor at 23.3 TB/s. Apply-pass x re-read should hit 192MB L2.
// Apply kernel stages its 64x128 x-tile via the Tensor Data Mover
// (tensor_load_to_lds + s_wait_tensorcnt).

typedef __attribute__((ext_vector_type(2))) float v2f;
typedef __attribute__((ext_vector_type(8))) float v8f;
typedef __attribute__((ext_vector_type(4))) unsigned int u32x4;
typedef __attribute__((ext_vector_type(8))) int i32x8;
typedef __attribute__((ext_vector_type(4))) int i32x4;

#define EPSV   1e-4f
#define NB     64
#define CC     512
#define HWSZ   1024
#define GRP    8
#define LDIM   64
#define MTOT   (NB * HWSZ)   // 65536

// ws layout (floats): sums[512] | mu[512] | cov(S)[32768] | sub[32768]
#define WS_SUMS 0
#define WS_MU   512
#define WS_COV  1024
#define WS_SUB  33792

// ---------------------------------------------------------------- zero ws
__global__ void zca_zero_kernel(float* __restrict__ p, int n) {
    int i = blockIdx.x * 256 + threadIdx.x;
    if (i < n) p[i] = 0.0f;
}

// ---------------------------------------------------------------- fused sums + second moment
// grid = 8 groups * 64 M-slices (slice == batch index n). 512 threads = 16 waves.
// One pass over x: WMMA accumulates S = sum_m x x^T per group (16x16 tile per
// wave via v_wmma_f32_16x16x4_f32 on LDS-staged tiles); threads 0..63 accumulate
// per-channel row sums. Split-K finished with global f32 atomics.
__global__ __launch_bounds__(512) void zca_cov_kernel(const float* __restrict__ x,
                                                      float* __restrict__ sums,
                                                      float* __restrict__ cov) {
    const int g     = blockIdx.x >> 6;
    const int slice = blockIdx.x & 63;          // == n
    const int tid   = threadIdx.x;
    const int lane  = tid & 31;
    const int wave  = tid >> 5;                 // 0..15
    const int half  = lane >> 4;                // 0/1
    const int r     = lane & 15;
    const int ti    = wave >> 2;                // row tile 0..3
    const int tj    = wave & 3;                 // col tile 0..3

    __shared__ float tile[64][68];              // 64 chans x 64 m, padded

    v8f acc = {};
    float rs = 0.0f;                            // row sum (threads 0..63)
    for (int mt = 0; mt < 16; ++mt) {           // 16 sub-tiles of 64 m each
        // coalesced float4 staging of raw x
        #pragma unroll
        for (int i = 0; i < 2; ++i) {
            int flat = tid + i * 512;           // 0..1023 float4 slots
            int row  = flat >> 4;
            int q4   = flat & 15;
            const float4 v = *(const float4*)(x +
                (((size_t)(slice * CC + g * 64 + row)) << 10) + mt * 64 + q4 * 4);
            tile[row][q4 * 4 + 0] = v.x;
            tile[row][q4 * 4 + 1] = v.y;
            tile[row][q4 * 4 + 2] = v.z;
            tile[row][q4 * 4 + 3] = v.w;
        }
        __syncthreads();
        if (tid < 64) {                         // per-channel sums for the mean
            #pragma unroll 8
            for (int q = 0; q < 64; ++q) rs += tile[tid][q];
        }
        #pragma unroll
        for (int k0 = 0; k0 < 64; k0 += 4) {
            v2f a, b;
            a.x = tile[ti * 16 + r][k0 + 2 * half];
            a.y = tile[ti * 16 + r][k0 + 2 * half + 1];
            b.x = tile[tj * 16 + r][k0 + 2 * half];
            b.y = tile[tj * 16 + r][k0 + 2 * half + 1];
            acc = __builtin_amdgcn_wmma_f32_16x16x4_f32(
                false, a, false, b, (short)0, acc, false, false);
        }
        __syncthreads();
    }
    if (tid < 64) atomicAdd(&sums[g * 64 + tid], rs);
    float* cg = cov + g * 4096;
    #pragma unroll
    for (int vv = 0; vv < 8; ++vv) {
        int row = ti * 16 + vv + half * 8;      // C/D layout: vgpr r -> M=r / M=r+8
        int col = tj * 16 + r;
        atomicAdd(&cg[row * 64 + col], acc[vv]);
    }
}

// ---------------------------------------------------------------- Jacobi eigensolver + whitening matrix
// One block per group, 64 threads. Finalizes mu and A = S/M - mu mu^T + eps*I,
// then parallel cyclic Jacobi (32 disjoint rotations/round, 63 rounds/sweep).
__global__ __launch_bounds__(64) void zca_eigen_kernel(const float* __restrict__ sums,
                                                       const float* __restrict__ cov,
                                                       float* __restrict__ mu,
                                                       float* __restrict__ sub) {
    const int g = blockIdx.x;
    const int t = threadIdx.x;                  // 0..63
    __shared__ float A[64][65];
    __shared__ float V[64][65];
    __shared__ float mmu[64];
    __shared__ float cs[32], sn[32];
    __shared__ int   top[32], bot[32];
    __shared__ float ev[64];
    __shared__ int   perm[64];
    __shared__ float wv[64];
    __shared__ int   counter;

    const float invM = 1.0f / (float)MTOT;
    float mu_t = sums[g * 64 + t] * invM;
    mmu[t] = mu_t;
    mu[g * 64 + t] = mu_t;
    if (t < 32) { top[t] = 2 * t; bot[t] = 2 * t + 1; }
    __syncthreads();

    for (int j = 0; j < 64; ++j) {
        float a = cov[g * 4096 + t * 64 + j] * invM - mu_t * mmu[j];
        if (j == t) a += EPSV;
        A[t][j] = a;
        V[t][j] = (j == t) ? 1.0f : 0.0f;
    }
    __syncthreads();

    for (int sw = 0; sw < 10; ++sw) {
        for (int round = 0; round < 63; ++round) {
            if (t < 32) {
                int p = top[t], q = bot[t];
                if (p > q) { int tm = p; p = q; q = tm; }
                float apq = A[p][q];
                float c = 1.0f, s = 0.0f;
                if (fabsf(apq) > 1e-12f) {
                    float theta = (A[q][q] - A[p][p]) / (2.0f * apq);
                    float tt = ((theta >= 0.0f) ? 1.0f : -1.0f) /
                               (fabsf(theta) + sqrtf(1.0f + theta * theta));
                    c = rsqrtf(1.0f + tt * tt);
                    s = tt * c;
                }
                cs[t] = c; sn[t] = s;
            }
            __syncthreads();
            // row updates: thread t owns column t (disjoint across threads)
            for (int j = 0; j < 32; ++j) {
                int p = top[j], q = bot[j];
                if (p > q) { int tm = p; p = q; q = tm; }
                float c = cs[j], s = sn[j];
                float ap = A[p][t], aq = A[q][t];
                A[p][t] = c * ap - s * aq;
                A[q][t] = s * ap + c * aq;
            }
            __syncthreads();
            // column + eigenvector updates: thread t owns row t
            for (int j = 0; j < 32; ++j) {
                int p = top[j], q = bot[j];
                if (p > q) { int tm = p; p = q; q = tm; }
                float c = cs[j], s = sn[j];
                float ap = A[t][p], aq = A[t][q];
                A[t][p] = c * ap - s * aq;
                A[t][q] = s * ap + c * aq;
                float vp = V[t][p], vq = V[t][q];
                V[t][p] = c * vp - s * vq;
                V[t][q] = s * vp + c * vq;
            }
            __syncthreads();
            if (t == 0) {                       // rotate tournament pairing
                int lastTop = top[31], b0 = bot[0];
                for (int i = 31; i >= 2; --i) top[i] = top[i - 1];
                top[1] = b0;
                for (int i = 0; i < 31; ++i) bot[i] = bot[i + 1];
                bot[31] = lastTop;
            }
            __syncthreads();
        }
    }

    ev[t]   = A[t][t];
    perm[t] = t;
    __syncthreads();
    // odd-even transposition sort, descending eigenvalues
    for (int ph = 0; ph < 64; ++ph) {
        if (t < 32) {
            int i = 2 * t + (ph & 1);
            if (i + 1 < 64) {
                int pi = perm[i], pj = perm[i + 1];
                if (ev[pi] < ev[pj]) { perm[i] = pj; perm[i + 1] = pi; }
            }
        }
        __syncthreads();
    }
    if (t == 0) {                               // faithful counter_i logic
        int cnt = 1;
        for (int j = 0; j < 64; ++j) {
            if (ev[perm[j]] <= EPSV) { cnt = j + 1; break; }
        }
        counter = cnt;
    }
    __syncthreads();
    wv[t] = (t < counter) ? rsqrtf(ev[perm[t]]) : 0.0f;
    __syncthreads();

    // sub[t][k] = sum_j wv[j] * V[t][perm[j]] * V[k][perm[j]]
    for (int k = 0; k < 64; ++k) {
        float a = 0.0f;
        for (int j = 0; j < 64; ++j) {
            float wj = wv[j];
            if (wj != 0.0f) {
                int pj = perm[j];
                a += wj * V[t][pj] * V[k][pj];
            }
        }
        sub[g * 4096 + t * 64 + k] = a;
    }
}

// ---------------------------------------------------------------- apply whitening
// grid = 8 groups * 512 m-chunks(128 wide). 256 threads = 8 waves.
// x-tile (64 rows x 128 cols, row stride 4KiB) staged to LDS by the Tensor
// Data Mover; LDS padding (4 DWORDs per 128) done by TDM pad_enable.
// y[c][m] = (sum_k sub[g][c%64][k] * (x[g*64+k][m]-mu)) * weight[c] + bias[c]
__global__ __launch_bounds__(256) void zca_apply_kernel(const float* __restrict__ x,
                                                        const float* __restrict__ mu,
                                                        const float* __restrict__ sub,
                                                        const float* __restrict__ weight,
                                                        const float* __restrict__ bias,
                                                        float* __restrict__ out) {
    const int g     = blockIdx.x >> 9;
    const int chunk = blockIdx.x & 511;
    const int m0    = chunk * 128;
    const int n     = m0 >> 10;                 // 128 | 1024 -> single n per chunk
    const int hw0   = m0 & 1023;
    const int tid   = threadIdx.x;
    const int lane  = tid & 31;
    const int wave  = tid >> 5;                 // 0..7
    const int half  = lane >> 4;
    const int r     = lane & 15;

    __shared__ float xcL[64][132];              // raw x tile: 64 k x 128 m (+4 pad)
    __shared__ float subL[64][68];
    __shared__ float mug[64];
    if (tid < 64) mug[tid] = mu[g * 64 + tid];

    #pragma unroll
    for (int i = 0; i < 4; ++i) {               // sub: 1024 float4 (L2-resident)
        int flat = tid + i * 256;
        int row = flat >> 4, q4 = flat & 15;
        const float4 v = *(const float4*)(sub + g * 4096 + row * 64 + q4 * 4);
        subL[row][q4 * 4 + 0] = v.x;
        subL[row][q4 * 4 + 1] = v.y;
        subL[row][q4 * 4 + 2] = v.z;
        subL[row][q4 * 4 + 3] = v.w;
    }

    if (wave == 0) {                            // TDM: one DMA for the whole tile
        uint64_t ga  = (uint64_t)(uintptr_t)(x +
            (((size_t)(n * CC + g * 64)) << 10) + hw0);
        uint32_t lds = (uint32_t)(uintptr_t)&xcL[0][0];
        u32x4 d0;
        d0[0] = 1u;                                             // count=1, user D#
        d0[1] = lds;                                            // lds_addr (bytes)
        d0[2] = (uint32_t)(ga & 0xFFFFFFFFu);                   // global_addr[31:0]
        d0[3] = (uint32_t)((ga >> 32) & 0x01FFFFFFu) | (2u << 30); // addr[56:32]|type=2
        i32x8 d1;
        d1[0] = (int)((2u << 16) |                              // data_size = 4B
                      (1u << 20) |                              // pad_enable
                      (6u << 22) |                              // pad_interval: 128 DW
                      (3u << 25));                              // pad_amount: 4 DW
        d1[1] = (int)(1024u << 16);                             // tensor_dim0 = 1024
        d1[2] = (int)(64u << 16);                               // tensor_dim1 = 64
        d1[3] = (int)(128u << 16);                              // tile_dim0 = 128
        d1[4] = 64;                                             // tile_dim1 = 64
        d1[5] = 1024;                                           // dim0_stride = 1024
        d1[6] = 0;
        d1[7] = 0;
        i32x4 d2 = {0, 0, 0, 0};
        i32x4 d3 = {0, 0, 0, 0};
        i32x8 d4 = {0, 0, 0, 0, 0, 0, 0, 0};    // extra group (6-arg builtin form)
        __builtin_amdgcn_tensor_load_to_lds(d0, d1, d2, d3, d4, 0);
        __builtin_amdgcn_s_wait_tensorcnt(0);
    }
    __syncthreads();

    const int mt = wave;                        // m sub-tile 0..7
    for (int ri = 0; ri < 4; ++ri) {            // row tile 0..3
        v8f acc = {};
        #pragma unroll
        for (int k0 = 0; k0 < 64; k0 += 4) {
            v2f a, b;
            a.x = subL[ri * 16 + r][k0 + 2 * half];
            a.y = subL[ri * 16 + r][k0 + 2 * half + 1];
            b.x = xcL[k0 + 2 * half][mt * 16 + r]     - mug[k0 + 2 * half];
            b.y = xcL[k0 + 2 * half + 1][mt * 16 + r] - mug[k0 + 2 * half + 1];
            acc = __builtin_amdgcn_wmma_f32_16x16x4_f32(
                false, a, false, b, (short)0, acc, false, false);
        }
        #pragma unroll
        for (int vv = 0; vv < 8; ++vv) {
            int rl   = ri * 16 + vv + half * 8;
            int c    = g * 64 + rl;
            int mcol = hw0 + mt * 16 + r;
            out[(((size_t)(n * CC + c)) << 10) + mcol] = acc[vv] * weight[c] + bias[c];
        }
    }
}

// ---------------------------------------------------------------- launch
extern "C" void kernel_launch(void* const* d_in, const int* in_sizes, int n_in,
                              void* d_out, int out_size, void* d_ws, size_t ws_size,
                              hipStream_t stream) {
    (void)in_sizes; (void)n_in; (void)out_size; (void)ws_size;
    const float* x      = (const float*)d_in[0];
    const float* weight = (const float*)d_in[1];
    const float* bias   = (const float*)d_in[2];
    float* out = (float*)d_out;
    float* ws  = (float*)d_ws;

    float* sums = ws + WS_SUMS;   // 512 floats
    float* mu   = ws + WS_MU;     // 512 floats
    float* cov  = ws + WS_COV;    // 8*64*64 = 32768 floats (raw second moment S)
    float* sub  = ws + WS_SUB;    // 32768 floats (total ws use: 264 KB)

    const int nzero = WS_SUB;     // sums + mu + cov regions
    zca_zero_kernel <<<(nzero + 255) / 256, 256, 0, stream>>>(ws, nzero);
    zca_cov_kernel  <<<512, 512, 0, stream>>>(x, sums, cov);  // 8 groups * 64 slices
    zca_eigen_kernel<<<8, 64, 0, stream>>>(sums, cov, mu, sub);
    zca_apply_kernel<<<4096, 256, 0, stream>>>(x, mu, sub, weight, bias, out);
}